// Model_35450660061466
// MI455X (gfx1250) — compile-verified
//
#include <hip/hip_runtime.h>
#include <math.h>

typedef float v2f __attribute__((ext_vector_type(2)));
typedef float v8f __attribute__((ext_vector_type(8)));

#define NB      256
#define T_IN    720
#define T_TOT   1440
#define C_CH    321
#define NFREQ   200            // CUT_FREQ
#define NOUT    400            // FREQ_OUT
#define N1      (2 * NFREQ)    // 400  (Re|Im of X)
#define N2      (2 * NOUT)     // 800  (Re|Im of out_freq)
#define M_TOTAL (NB * C_CH)    // 82176

#define BM 256                 // block rows  (8 waves x 32)
#define BN 80                  // block cols  (wave tile 32x80 -> 10 accums)
#define KC 16                  // K chunk
#define LSA 20                 // As row stride  (floats)  -> 4-bank blocks
#define LSB 20                 // Bst col stride (floats)

// ---------------------------------------------------------------------------
// Basis generation: B1 [720][400] forward DFT, B2 [800][720] inverse (last 720)
// ---------------------------------------------------------------------------
__global__ __launch_bounds__(256) void build_basis(float* __restrict__ B1,
                                                   float* __restrict__ B2) {
    const int idx = blockIdx.x * 256 + threadIdx.x;
    const int n1 = T_IN * N1;   // 288000
    const int n2 = N2 * T_IN;   // 576000
    if (idx < n1) {
        const int t = idx / N1;
        const int n = idx - t * N1;
        const int f = (n < NFREQ) ? n : n - NFREQ;
        const int p = (f * t) % T_IN;                       // exact phase
        const float th = (float)((double)p * (6.283185307179586 / (double)T_IN));
        B1[idx] = (n < NFREQ) ? cosf(th) : -sinf(th);
    }
    const int idx2 = idx - n1;
    if (idx2 >= 0 && idx2 < n2) {
        const int k = idx2 / T_IN;
        const int t = idx2 - k * T_IN;
        const int o = (k < NOUT) ? k : k - NOUT;
        const int tau = T_IN + t;                           // last 720 samples
        const int p = (o * tau) % T_TOT;                    // exact phase
        const float th = (float)((double)p * (6.283185307179586 / (double)T_TOT));
        // irfft scale 1/1440, factor 2 for o>=1 Hermitian fold, LENGTH_RATIO=2
        const float alpha = ((o == 0) ? 2.0f : 4.0f) / (float)T_TOT;
        B2[idx2] = (k < NOUT) ? alpha * cosf(th) : -alpha * sinf(th);
    }
}

// ---------------------------------------------------------------------------
// Per-(b,c) mean / stdev (ddof=1) / 1/stdev
// ---------------------------------------------------------------------------
__global__ __launch_bounds__(256) void stats_kernel(const float* __restrict__ x,
                                                    float* __restrict__ meanO,
                                                    float* __restrict__ stdO,
                                                    float* __restrict__ invstdO) {
    const int m = blockIdx.x * 256 + threadIdx.x;
    if (m >= M_TOTAL) return;
    const int b = m / C_CH;
    const int c = m - b * C_CH;
    const float* p = x + (size_t)b * T_IN * C_CH + c;
    float s = 0.0f, s2 = 0.0f;
    for (int t = 0; t < T_IN; ++t) {
        const float v = p[(size_t)t * C_CH];
        s += v; s2 += v * v;
    }
    const float mu  = s * (1.0f / (float)T_IN);
    const float var = (s2 - s * mu) * (1.0f / (float)(T_IN - 1));
    const float sd  = sqrtf(var + 1e-5f);
    meanO[m] = mu; stdO[m] = sd; invstdO[m] = 1.0f / sd;
}

// Per-chunk compute: 4 K-steps x (2 M x 5 N) = 40 WMMAs.
// As: [BM][LSA] row-major, Bst: [BN][LSB] col-major (k contiguous) -> both
// fragments are contiguous 8B-aligned pairs (single ds_load_b64 each).
#define WMMA_CHUNK(ASP, BSP)                                                    \
    _Pragma("unroll")                                                           \
    for (int kk = 0; kk < 16; kk += 4) {                                        \
        v2f a0 = *(const v2f*)&(ASP)[mrow0 * LSA + kk + khalf];                 \
        v2f a1 = *(const v2f*)&(ASP)[mrow1 * LSA + kk + khalf];                 \
        _Pragma("unroll")                                                       \
        for (int j = 0; j < 5; ++j) {                                           \
            v2f b = *(const v2f*)&(BSP)[(ncol + 16 * j) * LSB + kk + khalf];    \
            acc[0][j] = __builtin_amdgcn_wmma_f32_16x16x4_f32(                  \
                false, a0, false, b, (short)0, acc[0][j], false, false);        \
            acc[1][j] = __builtin_amdgcn_wmma_f32_16x16x4_f32(                  \
                false, a1, false, b, (short)0, acc[1][j], false, false);        \
        }                                                                       \
    }

// ---------------------------------------------------------------------------
// GEMM A: Y1[m, n] = sum_t ((x[b,t,c]-mu)/sd) * B1[t, n]
//   M = 82176 (m = b*321+c), K = 720, N = 400
// ---------------------------------------------------------------------------
__global__ __launch_bounds__(256) void gemm_dft(const float* __restrict__ x,
                                                const float* __restrict__ B1,
                                                const float* __restrict__ meanI,
                                                const float* __restrict__ invstdI,
                                                float* __restrict__ Y1) {
    __shared__ float As[2][BM * LSA];
    __shared__ float Bst[2][BN * LSB];
    const int tid  = threadIdx.x;
    const int lane = tid & 31;
    const int wave = tid >> 5;
    const int mBlock = blockIdx.x * BM;
    const int nBlock = blockIdx.y * BN;

    // A loader: one row per thread, 16 k per chunk, k staggered by tid>>4
    const int m  = mBlock + tid;
    const int bb = m / C_CH;
    const int cc = m - bb * C_CH;
    const float mu = meanI[m];
    const float is = invstdI[m];
    const float* arowp = x + (size_t)bb * T_IN * C_CH + cc;
    const int kst = tid >> 4;                      // write-stagger
    // B loader: kr = tid/16 (k row), n0 = tid%16 (+16j)
    const int kr = tid >> 4;
    const int n0 = tid & 15;

    const int mrow0 = (wave << 5) + (lane & 15);
    const int mrow1 = mrow0 + 16;
    const int khalf = (lane >> 4) << 1;            // 0 or 2
    const int ncol  = lane & 15;

    float aReg[16];
    float bReg[5];
    auto loadA = [&](int k0) {
#pragma unroll
        for (int i = 0; i < 16; ++i) {
            const int k = (i + kst) & 15;
            aReg[i] = arowp[(size_t)(k0 + k) * C_CH];
        }
    };
    auto loadB = [&](int k0) {
#pragma unroll
        for (int j = 0; j < 5; ++j)
            bReg[j] = B1[(size_t)(k0 + kr) * N1 + nBlock + n0 + 16 * j];
    };
    auto stage = [&](int buf) {
#pragma unroll
        for (int i = 0; i < 16; ++i) {
            const int k = (i + kst) & 15;
            As[buf][tid * LSA + k] = (aReg[i] - mu) * is;
        }
#pragma unroll
        for (int j = 0; j < 5; ++j)
            Bst[buf][(n0 + 16 * j) * LSB + kr] = bReg[j];
    };

    v8f acc[2][5] = {};
    const int nChunks = T_IN / KC;                 // 45
    loadA(0); loadB(0); stage(0);
    __syncthreads();
    for (int ch = 0; ch < nChunks; ++ch) {
        const int cur = ch & 1;
        const bool more = (ch + 1 < nChunks);
        if (more) { loadA((ch + 1) * KC); loadB((ch + 1) * KC); }
        WMMA_CHUNK(As[cur], Bst[cur])
        if (more) stage(cur ^ 1);
        __syncthreads();
    }
#pragma unroll
    for (int h = 0; h < 2; ++h) {
        const int mBase = mBlock + (wave << 5) + (h << 4) + ((lane >> 4) << 3);
#pragma unroll
        for (int j = 0; j < 5; ++j) {
            const int nCol = nBlock + 16 * j + ncol;
#pragma unroll
            for (int r = 0; r < 8; ++r)
                Y1[(size_t)(mBase + r) * N1 + nCol] = acc[h][j][r];
        }
    }
}

// ---------------------------------------------------------------------------
// GEMM B: per-channel complex linear + bias.
//   For channel c: Z[b, n] = sum_k A[b,k] * Bc[k,n] + bias
//   A[b,k] = Y1[(b*321+c)*400 + k]      (K = 400 = Re|Im of X)
//   Bc[k,n]: k<200 (f=k):     n<400 -> Wr[c,n,f]     n>=400 -> Wi[c,n-400,f]
//            k>=200 (f=k-200): n<400 -> -Wi[c,n,f]   n>=400 -> Wr[c,n-400,f]
//   Output Y2[(b*321+c)*800 + n], N = 800 (Re|Im of out_freq)
// ---------------------------------------------------------------------------
__global__ __launch_bounds__(256) void gemm_cmul(const float* __restrict__ Y1,
                                                 const float* __restrict__ Wr,
                                                 const float* __restrict__ Wi,
                                                 const float* __restrict__ br,
                                                 const float* __restrict__ bi,
                                                 float* __restrict__ Y2) {
    __shared__ float As[2][BM * LSA];
    __shared__ float Bst[2][BN * LSB];
    const int tid  = threadIdx.x;
    const int lane = tid & 31;
    const int wave = tid >> 5;
    const int nBlock = blockIdx.y * BN;   // 0..720 step 80
    const int c = blockIdx.z;

    // A loader: ks = tid%16 (k contiguous), 16 rows/pass x 16 passes
    const int ks = tid & 15;
    const int rg = tid >> 4;
    // B loader: bkk = tid%16 (f contiguous), col = tid/16 (+16j)
    const int bkk = tid & 15;
    const int bn0 = tid >> 4;

    const int mrow0 = (wave << 5) + (lane & 15);
    const int mrow1 = mrow0 + 16;
    const int khalf = (lane >> 4) << 1;
    const int ncol  = lane & 15;

    float aReg[16];
    float bReg[5];
    auto loadA = [&](int k0) {
#pragma unroll
        for (int p = 0; p < 16; ++p) {
            const int b = rg + p * 16;            // BM == NB == 256
            aReg[p] = Y1[((size_t)b * C_CH + c) * N1 + (k0 + ks)];
        }
    };
    auto loadB = [&](int k0) {
        const int kg = k0 + bkk;
        const int f = (kg < NFREQ) ? kg : kg - NFREQ;
#pragma unroll
        for (int j = 0; j < 5; ++j) {
            const int ng = nBlock + bn0 + 16 * j;
            const int o = (ng < NOUT) ? ng : ng - NOUT;
            const float* Wsrc;
            float sgn = 1.0f;
            if (kg < NFREQ) { Wsrc = (ng < NOUT) ? Wr : Wi; }
            else { if (ng < NOUT) { Wsrc = Wi; sgn = -1.0f; } else { Wsrc = Wr; } }
            bReg[j] = sgn * Wsrc[((size_t)c * NOUT + o) * NFREQ + f];
        }
    };
    auto stage = [&](int buf) {
#pragma unroll
        for (int p = 0; p < 16; ++p)
            As[buf][(rg + p * 16) * LSA + ks] = aReg[p];
#pragma unroll
        for (int j = 0; j < 5; ++j)
            Bst[buf][(bn0 + 16 * j) * LSB + bkk] = bReg[j];
    };

    v8f acc[2][5] = {};
    const int nChunks = N1 / KC;          // 25
    loadA(0); loadB(0); stage(0);
    __syncthreads();
    for (int ch = 0; ch < nChunks; ++ch) {
        const int cur = ch & 1;
        const bool more = (ch + 1 < nChunks);
        if (more) { loadA((ch + 1) * KC); loadB((ch + 1) * KC); }
        WMMA_CHUNK(As[cur], Bst[cur])
        if (more) stage(cur ^ 1);
        __syncthreads();
    }
#pragma unroll
    for (int h = 0; h < 2; ++h) {
        const int mBase = (wave << 5) + (h << 4) + ((lane >> 4) << 3);
#pragma unroll
        for (int j = 0; j < 5; ++j) {
            const int nCol = nBlock + 16 * j + ncol;
            const int o = (nCol < NOUT) ? nCol : nCol - NOUT;
            const float bias = (nCol < NOUT) ? br[(size_t)c * NOUT + o]
                                             : bi[(size_t)c * NOUT + o];
#pragma unroll
            for (int r = 0; r < 8; ++r) {
                const int b = mBase + r;
                Y2[((size_t)b * C_CH + c) * N2 + nCol] = acc[h][j][r] + bias;
            }
        }
    }
}

// ---------------------------------------------------------------------------
// GEMM C: out[b, t, c] = (sum_k Y2[m,k] * B2[k,t]) * sd[m] + mu[m]
//   M = 82176, K = 800, N = 720
// ---------------------------------------------------------------------------
__global__ __launch_bounds__(256) void gemm_irfft(const float* __restrict__ Y2,
                                                  const float* __restrict__ B2,
                                                  const float* __restrict__ meanI,
                                                  const float* __restrict__ stdI,
                                                  float* __restrict__ out) {
    __shared__ float As[2][BM * LSA];
    __shared__ float Bst[2][BN * LSB];
    const int tid  = threadIdx.x;
    const int lane = tid & 31;
    const int wave = tid >> 5;
    const int mBlock = blockIdx.x * BM;
    const int nBlock = blockIdx.y * BN;   // t offset, 0..640 step 80

    const int ks = tid & 15;
    const int rg = tid >> 4;
    const int kr = tid >> 4;
    const int n0 = tid & 15;

    const int mrow0 = (wave << 5) + (lane & 15);
    const int mrow1 = mrow0 + 16;
    const int khalf = (lane >> 4) << 1;
    const int ncol  = lane & 15;

    float aReg[16];
    float bReg[5];
    auto loadA = [&](int k0) {
#pragma unroll
        for (int p = 0; p < 16; ++p)
            aReg[p] = Y2[(size_t)(mBlock + rg + p * 16) * N2 + (k0 + ks)];
    };
    auto loadB = [&](int k0) {
#pragma unroll
        for (int j = 0; j < 5; ++j)
            bReg[j] = B2[(size_t)(k0 + kr) * T_IN + nBlock + n0 + 16 * j];
    };
    auto stage = [&](int buf) {
#pragma unroll
        for (int p = 0; p < 16; ++p)
            As[buf][(rg + p * 16) * LSA + ks] = aReg[p];
#pragma unroll
        for (int j = 0; j < 5; ++j)
            Bst[buf][(n0 + 16 * j) * LSB + kr] = bReg[j];
    };

    v8f acc[2][5] = {};
    const int nChunks = N2 / KC;          // 50
    loadA(0); loadB(0); stage(0);
    __syncthreads();
    for (int ch = 0; ch < nChunks; ++ch) {
        const int cur = ch & 1;
        const bool more = (ch + 1 < nChunks);
        if (more) { loadA((ch + 1) * KC); loadB((ch + 1) * KC); }
        WMMA_CHUNK(As[cur], Bst[cur])
        if (more) stage(cur ^ 1);
        __syncthreads();
    }
#pragma unroll
    for (int h = 0; h < 2; ++h) {
        const int mBase = mBlock + (wave << 5) + (h << 4) + ((lane >> 4) << 3);
#pragma unroll
        for (int r = 0; r < 8; ++r) {
            const int m = mBase + r;
            const int b = m / C_CH;
            const int c = m - b * C_CH;
            const float sd = stdI[m];
            const float muv = meanI[m];
            float* orow = out + (size_t)b * T_IN * C_CH + c;
#pragma unroll
            for (int j = 0; j < 5; ++j) {
                const int tCol = nBlock + 16 * j + ncol;
                orow[(size_t)tCol * C_CH] = acc[h][j][r] * sd + muv;
            }
        }
    }
}

// ---------------------------------------------------------------------------
extern "C" void kernel_launch(void* const* d_in, const int* in_sizes, int n_in,
                              void* d_out, int out_size, void* d_ws, size_t ws_size,
                              hipStream_t stream) {
    (void)in_sizes; (void)n_in; (void)out_size; (void)ws_size;
    const float* x_enc = (const float*)d_in[0];
    const float* Wr    = (const float*)d_in[4];
    const float* Wi    = (const float*)d_in[5];
    const float* br    = (const float*)d_in[6];
    const float* bi    = (const float*)d_in[7];
    float* out = (float*)d_out;

    char* ws = (char*)d_ws;
    size_t off = 0;
    auto take = [&](size_t bytes) -> float* {
        float* p = (float*)(ws + off);
        off = (off + bytes + 255) & ~(size_t)255;
        return p;
    };
    float* B1     = take((size_t)T_IN * N1 * sizeof(float));      //  1.15 MB
    float* B2     = take((size_t)N2 * T_IN * sizeof(float));      //  2.30 MB
    float* meanW  = take((size_t)M_TOTAL * sizeof(float));
    float* stdW   = take((size_t)M_TOTAL * sizeof(float));
    float* invstd = take((size_t)M_TOTAL * sizeof(float));
    float* Y1     = take((size_t)M_TOTAL * N1 * sizeof(float));   // 131.5 MB
    float* Y2     = take((size_t)M_TOTAL * N2 * sizeof(float));   // 263.0 MB

    {
        const int total = T_IN * N1 + N2 * T_IN;   // 864000
        build_basis<<<(total + 255) / 256, 256, 0, stream>>>(B1, B2);
    }
    stats_kernel<<<(M_TOTAL + 255) / 256, 256, 0, stream>>>(x_enc, meanW, stdW, invstd);

    gemm_dft<<<dim3(M_TOTAL / BM, N1 / BN), 256, 0, stream>>>(x_enc, B1, meanW, invstd, Y1);

    gemm_cmul<<<dim3(NB / BM, N2 / BN, C_CH), 256, 0, stream>>>(Y1, Wr, Wi, br, bi, Y2);

    gemm_irfft<<<dim3(M_TOTAL / BM, T_IN / BN), 256, 0, stream>>>(Y2, B2, meanW, stdW, out);
}